// BilinearJoints_81621558493836
// MI455X (gfx1250) — compile-verified
//
#include <hip/hip_runtime.h>

#define DT_F    0.0166667f
#define BATCH   16384
#define NJ      64
#define NM      8
#define NELEM   (BATCH * NJ)
#define TPB     256

typedef __attribute__((ext_vector_type(4))) unsigned int u32x4;
typedef __attribute__((ext_vector_type(8))) int          i32x8;
typedef __attribute__((ext_vector_type(4))) int          i32x4;
typedef __attribute__((ext_vector_type(2))) float        f32x2;
typedef __attribute__((ext_vector_type(4))) float        f32x4;

#if defined(__HIP_DEVICE_COMPILE__) && __has_builtin(__builtin_amdgcn_tensor_load_to_lds)
#define HAVE_TDM 1
#else
#define HAVE_TDM 0
#endif

#if HAVE_TDM
// One-shot 1-D DMA copy of n float32 from global to LDS via the Tensor Data
// Mover (CDNA5 TENSOR_LOAD_TO_LDS). Descriptor layout per CDNA5 ISA.
__device__ __forceinline__ void tdm_load_1d(const float* gptr, float* lptr, unsigned n) {
    unsigned long long ga = (unsigned long long)gptr;
    unsigned la = (unsigned)(unsigned long long)lptr;   // low 32 bits = LDS byte offset

    u32x4 g0;
    g0.x = 1u;                                     // count=1, user descriptor
    g0.y = la;                                     // lds_addr
    g0.z = (unsigned)(ga & 0xFFFFFFFFull);         // global_addr[31:0]
    g0.w = ((unsigned)(ga >> 32) & 0x01FFFFFFu)    // global_addr[56:32]
         | (2u << 30);                             // type = 2 ("image")

    i32x8 g1;
    g1[0] = 0x00020000;                            // data_size=2 (4 bytes/elem)
    g1[1] = (int)((n & 0xFFFFu) << 16);            // tensor_dim0[15:0]
    g1[2] = (int)((n >> 16) | (1u << 16));         // tensor_dim0[31:16] | tensor_dim1=1
    g1[3] = (int)((n & 0xFFFFu) << 16);            // tile_dim0 = n
    g1[4] = 1;                                     // tile_dim1=1, tile_dim2=0
    g1[5] = (int)n;                                // tensor_dim0_stride[31:0]
    g1[6] = (int)((n & 0xFFFFu) << 16);            // stride hi=0 | tensor_dim1_stride lo
    g1[7] = 0;

    i32x4 gz = {0, 0, 0, 0};                       // groups 2/3 unused (<=2-D tensor)
#if defined(__clang_major__) && (__clang_major__ >= 23)
    i32x8 z8 = {0, 0, 0, 0, 0, 0, 0, 0};
    __builtin_amdgcn_tensor_load_to_lds(g0, g1, gz, gz, z8, 0);
#else
    __builtin_amdgcn_tensor_load_to_lds(g0, g1, gz, gz, 0);
#endif
}
#endif

#if defined(__HIP_DEVICE_COMPILE__)
#define FMED3(x, lo, hi) __builtin_amdgcn_fmed3f((x), (lo), (hi))
#define FRCP(x)          __builtin_amdgcn_rcpf((x))
#define FSQRT(x)         __builtin_amdgcn_sqrtf((x))
#else
#define FMED3(x, lo, hi) fminf(fmaxf((x), (lo)), (hi))
#define FRCP(x)          (1.0f / (x))
#define FSQRT(x)         sqrtf((x))
#endif

__global__ __launch_bounds__(TPB) void bilinear_joints_kernel(
    const float* __restrict__ SS,   const float* __restrict__ Alphas,
    const float* __restrict__ K0s,  const float* __restrict__ K1s,
    const float* __restrict__ L0s,  const float* __restrict__ L1s,
    const float* __restrict__ Ms,   const float* __restrict__ Iv,
    const float* __restrict__ Bj,   const float* __restrict__ Kj,
    float* __restrict__ out0, float* __restrict__ out1)
{
    // raw parameter tables (TDM-staged), then per-joint derived tables
    __shared__ float rK0[NJ * NM], rK1[NJ * NM], rL0[NJ * NM], rL1[NJ * NM], rMm[NJ * NM];
    __shared__ float rI[NJ], rBj[NJ], rKj[NJ];
    __shared__ float sd1[NJ * NM], sc1[NJ * NM], sc2[NJ * NM];   // K1*M^2, (K0L1+K1L0)*M, 2*K1*L1*M
    __shared__ float sJ[NJ * 8];                                 // {invI, Kj/I, Bj/I, D0, C0, -, -, -}

#if HAVE_TDM
    if (threadIdx.x < 32) {   // wave 0 issues the DMA descriptors
        tdm_load_1d(K0s, rK0, NJ * NM);
        tdm_load_1d(K1s, rK1, NJ * NM);
        tdm_load_1d(L0s, rL0, NJ * NM);
        tdm_load_1d(L1s, rL1, NJ * NM);
        tdm_load_1d(Ms,  rMm, NJ * NM);
        tdm_load_1d(Iv,  rI,  NJ);
        tdm_load_1d(Bj,  rBj, NJ);
        tdm_load_1d(Kj,  rKj, NJ);
        __builtin_amdgcn_s_wait_tensorcnt(0);
    }
#else
    for (int i = threadIdx.x; i < NJ * NM; i += TPB) {
        rK0[i] = K0s[i]; rK1[i] = K1s[i]; rL0[i] = L0s[i]; rL1[i] = L1s[i]; rMm[i] = Ms[i];
    }
    for (int i = threadIdx.x; i < NJ; i += TPB) { rI[i] = Iv[i]; rBj[i] = Bj[i]; rKj[i] = Kj[i]; }
#endif
    __syncthreads();

    // Derive alpha-independent per-(joint,muscle) and per-joint constants (once per block).
    for (int i = threadIdx.x; i < NJ * NM; i += TPB) {
        const float K0 = rK0[i], K1 = rK1[i], L0 = rL0[i], L1 = rL1[i], M = rMm[i];
        sd1[i] = K1 * M * M;
        sc1[i] = fmaf(K0, L1, K1 * L0) * M;
        sc2[i] = 2.0f * K1 * L1 * M;
    }
    if (threadIdx.x < NJ) {
        const int j = threadIdx.x;
        float D0 = 0.0f, C0 = 0.0f;
#pragma unroll
        for (int m = 0; m < NM; ++m) {
            const float K0 = rK0[j * NM + m], L0 = rL0[j * NM + m], M = rMm[j * NM + m];
            D0 = fmaf(K0, M * M, D0);          // sum K0*M^2
            C0 = fmaf(K0 * L0, M, C0);         // sum K0*L0*M
        }
        const float invI = FRCP(rI[j]);
        sJ[j * 8 + 0] = invI;
        sJ[j * 8 + 1] = rKj[j] * invI;
        sJ[j * 8 + 2] = rBj[j] * invI;
        sJ[j * 8 + 3] = D0;
        sJ[j * 8 + 4] = C0;
    }
    __syncthreads();

    const int e = blockIdx.x * TPB + threadIdx.x;   // element = (batch, joint)
    const int j = e & (NJ - 1);

    // streamed inputs (read exactly once -> non-temporal)
    const f32x2 ss = __builtin_nontemporal_load((const f32x2*)SS + e);
    const f32x4 a0 = __builtin_nontemporal_load((const f32x4*)Alphas + 2 * e);
    const f32x4 a1 = __builtin_nontemporal_load((const f32x4*)Alphas + 2 * e + 1);
    const float al[NM] = {a0.x, a0.y, a0.z, a0.w, a1.x, a1.y, a1.z, a1.w};

    // per-joint derived tables from LDS (ds_load_b128 pairs)
    float d1a[NM], c1a[NM], c2a[NM];
    *(f32x4*)(d1a)     = ((const f32x4*)(sd1 + j * NM))[0];
    *(f32x4*)(d1a + 4) = ((const f32x4*)(sd1 + j * NM))[1];
    *(f32x4*)(c1a)     = ((const f32x4*)(sc1 + j * NM))[0];
    *(f32x4*)(c1a + 4) = ((const f32x4*)(sc1 + j * NM))[1];
    *(f32x4*)(c2a)     = ((const f32x4*)(sc2 + j * NM))[0];
    *(f32x4*)(c2a + 4) = ((const f32x4*)(sc2 + j * NM))[1];
    const f32x4 jr = *(const f32x4*)(sJ + j * 8);         // invI, Kj/I, Bj/I, D0
    const float C0 = sJ[j * 8 + 4];
    const float invI = jr.x, KjI = jr.y, BjI = jr.z, D0 = jr.w;

    float Kv = 0.0f, BF = 0.0f;
#pragma unroll
    for (int m = 0; m < NM; ++m) {
        const float a = FMED3(al[m], 0.0f, 1.0f);         // clip(alpha, 0, 1)
        Kv = fmaf(d1a[m], a, Kv);                         // variable part of Ksum
        BF = fmaf(c1a[m], a, BF);
        BF = fmaf(c2a[m], a * a, BF);
    }
    const float Ksum = Kv + D0;
    BF = C0 + BF - Ksum * fabsf(ss.x);                    // - sum(Kact)*|ss0|

    const float P    = fmaf(Ksum, invI, KjI);             // P = -A10 > 0
    const float q    = FSQRT(Ksum * invI);                // sqrt(Ksum/I)
    const float A11n = fmaf(2.0f, q, BjI);                // -A11 > 0
    const float B10  = BF * invI;

    // Closed-form expm of X = dt*A, A = [[0,1],[-P,-A11n]]
    // e^X = e^mu*(C*I + S*(X - mu*I));  mu = -A11n*dt/2;  Delta = mu^2 - P*dt^2
    // |Delta| <= ~1.5e-3 and |mu| <= ~0.021 for this problem's parameter ranges,
    // so short even/odd series are exact to fp32 — branch-free, no transcendentals.
    const float mu    = -0.5f * DT_F * A11n;
    const float Delta = fmaf(mu, mu, -(DT_F * DT_F) * P);
    const float C  = fmaf(Delta, fmaf(Delta, 1.0f / 24.0f, 0.5f), 1.0f);         // cosh(s)
    const float S  = fmaf(Delta, fmaf(Delta, 1.0f / 120.0f, 1.0f / 6.0f), 1.0f); // sinh(s)/s
    float h  = fmaf(mu, 1.0f / 24.0f, 1.0f / 6.0f);                              // e^mu (deg-4)
    h        = fmaf(mu, h, 0.5f);
    h        = fmaf(mu, h, 1.0f);
    const float em = fmaf(mu, h, 1.0f);

    const float t    = mu * S;
    const float Ad00 = em * (C - t);
    const float Ad11 = em * (C + t);
    const float Ad01 = em * S * DT_F;
    const float Ad10 = -Ad01 * P;

    // v = phi1(X)*(B*dt)*[1,0]^T = X^{-1}(e^X - I)[0, B10*dt]^T
    const float v1 = Ad01 * B10;
    const float v0 = B10 * fmaf(-A11n, Ad01, 1.0f - Ad11) * FRCP(P);

    const float o0 = fmaf(Ad00, ss.x, fmaf(Ad01, ss.y, v0));
    const float o1 = fmaf(Ad10, ss.x, fmaf(Ad11, ss.y, v1));

    f32x2 ov;
    ov.x = o0;
    ov.y = o1;
    __builtin_nontemporal_store(o0, out0 + e);
    __builtin_nontemporal_store(ov, (f32x2*)out1 + e);
}

extern "C" void kernel_launch(void* const* d_in, const int* in_sizes, int n_in,
                              void* d_out, int out_size, void* d_ws, size_t ws_size,
                              hipStream_t stream) {
    (void)in_sizes; (void)n_in; (void)out_size; (void)d_ws; (void)ws_size;
    const float* SS     = (const float*)d_in[0];
    const float* Alphas = (const float*)d_in[1];
    const float* K0s    = (const float*)d_in[2];
    const float* K1s    = (const float*)d_in[3];
    const float* L0s    = (const float*)d_in[4];
    const float* L1s    = (const float*)d_in[5];
    const float* Ms     = (const float*)d_in[6];
    const float* Iv     = (const float*)d_in[7];
    const float* Bj     = (const float*)d_in[8];
    const float* Kj     = (const float*)d_in[9];
    float* out0 = (float*)d_out;           // SSout[:, :, 0]  (NELEM floats)
    float* out1 = out0 + NELEM;            // SSout           (NELEM*2 floats)

    hipLaunchKernelGGL(bilinear_joints_kernel,
                       dim3(NELEM / TPB), dim3(TPB), 0, stream,
                       SS, Alphas, K0s, K1s, L0s, L1s, Ms, Iv, Bj, Kj, out0, out1);
}